// HeteroGraphConvLayer_41205916238557
// MI455X (gfx1250) — compile-verified
//
#include <hip/hip_runtime.h>

// ---------------------------------------------------------------------------
// HeteroGraphConv (user/item graph), D=128.
//   out = (cnt>0) ? W * (segment_sum(feat[src]) / cnt) + b : 0
// Linearity => aggregate raw features, project once per destination node:
//   W*(agg/cnt) row m = (1/cnt[m]) * (W*agg) row m,  agg row == 0 when cnt==0.
// Phase 1: zero scratch.
// Phase 2: per-edge atomic scatter of raw features (coalesced 512B rows/wave).
// Phase 3: WMMA f32 16x16x4 GEMM; A tile staged in LDS (shared by 8 waves),
//          1/cnt + masked bias applied in the epilogue.
// ---------------------------------------------------------------------------

#define N_USER 50000
#define N_ITEM 50000
#define DFEAT 128
#define LDA 132   // 128 + 4 pad: LDS bank stride 4 -> conflict-free A fragments

typedef float v2f __attribute__((ext_vector_type(2)));
typedef float v8f __attribute__((ext_vector_type(8)));

// ---------------------------------------------------------------- zero scratch
__global__ void hg_zero_f32(float* __restrict__ p, int n) {
    int i = blockIdx.x * blockDim.x + threadIdx.x;
    int stride = gridDim.x * blockDim.x;
    for (; i < n; i += stride) p[i] = 0.0f;
}

// ------------------------------------------------------- edge scatter (sum/cnt)
__global__ void hg_scatter(const float* __restrict__ feat,
                           const int* __restrict__ src,
                           const int* __restrict__ dst,
                           float* __restrict__ agg,
                           float* __restrict__ cnt,
                           int n_edges) {
    const int lane = threadIdx.x & 31;
    const int wave = threadIdx.x >> 5;
    const int e = blockIdx.x * (blockDim.x >> 5) + wave;
    if (e >= n_edges) return;
    const int s = src[e];
    const int d = dst[e];
    const float4* frow = (const float4*)(feat + (size_t)s * DFEAT);
    float* arow = agg + (size_t)d * DFEAT;
    float4 v = frow[lane];
    atomicAdd(&arow[lane * 4 + 0], v.x);
    atomicAdd(&arow[lane * 4 + 1], v.y);
    atomicAdd(&arow[lane * 4 + 2], v.z);
    atomicAdd(&arow[lane * 4 + 3], v.w);
    if (lane == 0) atomicAdd(&cnt[d], 1.0f);
}

// --------------------------------------------- cooperative global->LDS A tile
// 16 rows x 128 floats, written padded (LDA=132). 256 threads x 2 float4 each.
// Row byte offset m*528 is 16B aligned -> ds_store_b128.
__device__ __forceinline__ void hg_load_tile(float* __restrict__ As,
                                             const float* __restrict__ agg,
                                             int node0, int t) {
#pragma unroll
    for (int j = 0; j < 2; ++j) {
        const int e4 = t * 2 + j;        // 0..511 (512 float4 = 16x128 floats)
        const int m  = e4 >> 5;          // 32 float4 per row
        const int k4 = e4 & 31;
        float4 v = ((const float4*)(agg + (size_t)(node0 + m) * DFEAT))[k4];
        ((float4*)(As + m * LDA))[k4] = v;
    }
}

// ------------------------------------------------- WMMA 16x16 tile from LDS A
// V_WMMA_F32_16X16X4_F32 fragment layouts (wave32):
//   A 16x4 : lanes 0-15 rows M, K={k0,k0+1}; lanes 16-31 rows M, K={k0+2,k0+3}
//   B 4x16 : same K split, col N = lane%16
//   C 16x16: vgpr i -> row i (lanes 0-15) / i+8 (lanes 16-31), col = lane%16
__device__ __forceinline__ v8f hg_wmma_tile(const float* __restrict__ As,
                                            const float* __restrict__ W,
                                            int n, int lane) {
    const int m    = lane & 15;
    const int koff = (lane >> 4) * 2;    // 0 or 2
    const float* arow = As + m * LDA;
    v8f c0 = {};
    v8f c1 = {};
#pragma unroll
    for (int k0 = 0; k0 < DFEAT; k0 += 8) {
        {
            const int kb = k0 + koff;
            v2f a; a.x = arow[kb];                      a.y = arow[kb + 1];
            v2f b; b.x = W[(size_t)kb * DFEAT + n];     b.y = W[(size_t)(kb + 1) * DFEAT + n];
            c0 = __builtin_amdgcn_wmma_f32_16x16x4_f32(false, a, false, b,
                                                       (short)0, c0, false, false);
        }
        {
            const int kb = k0 + 4 + koff;               // independent accumulator
            v2f a; a.x = arow[kb];                      a.y = arow[kb + 1];
            v2f b; b.x = W[(size_t)kb * DFEAT + n];     b.y = W[(size_t)(kb + 1) * DFEAT + n];
            c1 = __builtin_amdgcn_wmma_f32_16x16x4_f32(false, a, false, b,
                                                       (short)0, c1, false, false);
        }
    }
#pragma unroll
    for (int i = 0; i < 8; ++i) c0[i] += c1[i];
    return c0;
}

// ------------------------------------------------------- user-node projection
// out_user = inv_uu*(W_uu.agg_uu) + mask_uu*b_uu + inv_iu*(W_iu.agg_iu) + mask_iu*b_iu
__global__ void hg_project_users(const float* __restrict__ agg_uu,
                                 const float* __restrict__ cnt_uu,
                                 const float* __restrict__ agg_iu,
                                 const float* __restrict__ cnt_iu,
                                 const float* __restrict__ W_uu,
                                 const float* __restrict__ b_uu,
                                 const float* __restrict__ W_iu,
                                 const float* __restrict__ b_iu,
                                 float* __restrict__ out_user) {
    __shared__ float As_uu[16 * LDA];
    __shared__ float As_iu[16 * LDA];
    __shared__ float inv_uu_s[16];
    __shared__ float inv_iu_s[16];

    const int t     = threadIdx.x;
    const int lane  = t & 31;
    const int wave  = t >> 5;            // 0..7 -> feature tile
    const int node0 = blockIdx.x * 16;
    const int n     = wave * 16 + (lane & 15);

    hg_load_tile(As_uu, agg_uu, node0, t);
    hg_load_tile(As_iu, agg_iu, node0, t);
    if (t < 16) {
        const float c = cnt_uu[node0 + t];
        inv_uu_s[t] = (c > 0.0f) ? (1.0f / c) : 0.0f;
    } else if (t < 32) {
        const float c = cnt_iu[node0 + t - 16];
        inv_iu_s[t - 16] = (c > 0.0f) ? (1.0f / c) : 0.0f;
    }
    __syncthreads();                     // full EXEC restored before WMMA

    v8f c_u = hg_wmma_tile(As_uu, W_uu, n, lane);
    v8f c_i = hg_wmma_tile(As_iu, W_iu, n, lane);

    const float bu = b_uu[n];
    const float bi = b_iu[n];
    const int rbase = (lane >> 4) * 8;
#pragma unroll
    for (int i = 0; i < 8; ++i) {
        const int row = rbase + i;
        const float iu = inv_uu_s[row];
        const float ii = inv_iu_s[row];
        float r = c_u[i] * iu + c_i[i] * ii;   // cnt==0 rows give exactly 0
        if (iu > 0.0f) r += bu;
        if (ii > 0.0f) r += bi;
        out_user[(size_t)(node0 + row) * DFEAT + n] = r;
    }
}

// ------------------------------------------------------- item-node projection
__global__ void hg_project_items(const float* __restrict__ agg_ub,
                                 const float* __restrict__ cnt_ub,
                                 const float* __restrict__ W_ub,
                                 const float* __restrict__ b_ub,
                                 float* __restrict__ out_item) {
    __shared__ float As_ub[16 * LDA];
    __shared__ float inv_ub_s[16];

    const int t     = threadIdx.x;
    const int lane  = t & 31;
    const int wave  = t >> 5;
    const int node0 = blockIdx.x * 16;
    const int n     = wave * 16 + (lane & 15);

    hg_load_tile(As_ub, agg_ub, node0, t);
    if (t < 16) {
        const float c = cnt_ub[node0 + t];
        inv_ub_s[t] = (c > 0.0f) ? (1.0f / c) : 0.0f;
    }
    __syncthreads();

    v8f c_b = hg_wmma_tile(As_ub, W_ub, n, lane);

    const float bb = b_ub[n];
    const int rbase = (lane >> 4) * 8;
#pragma unroll
    for (int i = 0; i < 8; ++i) {
        const int row = rbase + i;
        const float ib = inv_ub_s[row];
        float r = c_b[i] * ib;
        if (ib > 0.0f) r += bb;
        out_item[(size_t)(node0 + row) * DFEAT + n] = r;
    }
}

// ---------------------------------------------------------------------------
extern "C" void kernel_launch(void* const* d_in, const int* in_sizes, int n_in,
                              void* d_out, int out_size, void* d_ws, size_t ws_size,
                              hipStream_t stream) {
    const float* feat_user = (const float*)d_in[0];
    const float* feat_item = (const float*)d_in[1];
    const float* W_uu = (const float*)d_in[2];
    const float* b_uu = (const float*)d_in[3];
    const float* W_ub = (const float*)d_in[4];
    const float* b_ub = (const float*)d_in[5];
    const float* W_iu = (const float*)d_in[6];
    const float* b_iu = (const float*)d_in[7];
    const int* src_uu = (const int*)d_in[8];
    const int* dst_uu = (const int*)d_in[9];
    const int* src_ub = (const int*)d_in[10];
    const int* dst_ub = (const int*)d_in[11];
    const int* src_iu = (const int*)d_in[12];
    const int* dst_iu = (const int*)d_in[13];
    const int n_edge = in_sizes[8];

    // Workspace layout (floats): agg_uu | agg_ub | agg_iu | cnt_uu | cnt_ub | cnt_iu
    float* ws     = (float*)d_ws;
    float* agg_uu = ws;
    float* agg_ub = agg_uu + (size_t)N_USER * DFEAT;
    float* agg_iu = agg_ub + (size_t)N_ITEM * DFEAT;
    float* cnt_uu = agg_iu + (size_t)N_USER * DFEAT;
    float* cnt_ub = cnt_uu + N_USER;
    float* cnt_iu = cnt_ub + N_ITEM;
    const int total_ws_f32 = 2 * N_USER * DFEAT + N_ITEM * DFEAT + 2 * N_USER + N_ITEM;

    hg_zero_f32<<<2048, 256, 0, stream>>>(ws, total_ws_f32);

    const int wavesPerBlock = 8;  // 256 threads
    const int scatterBlocks = (n_edge + wavesPerBlock - 1) / wavesPerBlock;
    hg_scatter<<<scatterBlocks, 256, 0, stream>>>(feat_user, src_uu, dst_uu,
                                                  agg_uu, cnt_uu, n_edge);
    hg_scatter<<<scatterBlocks, 256, 0, stream>>>(feat_user, src_ub, dst_ub,
                                                  agg_ub, cnt_ub, n_edge);
    hg_scatter<<<scatterBlocks, 256, 0, stream>>>(feat_item, src_iu, dst_iu,
                                                  agg_iu, cnt_iu, n_edge);

    float* out_user = (float*)d_out;
    float* out_item = out_user + (size_t)N_USER * DFEAT;

    hg_project_users<<<N_USER / 16, 256, 0, stream>>>(agg_uu, cnt_uu, agg_iu, cnt_iu,
                                                      W_uu, b_uu, W_iu, b_iu, out_user);
    hg_project_items<<<N_ITEM / 16, 256, 0, stream>>>(agg_ub, cnt_ub, W_ub, b_ub,
                                                      out_item);
}